// ShapePriorLoss_31928786878945
// MI455X (gfx1250) — compile-verified
//
#include <hip/hip_runtime.h>
#include <hip/hip_bf16.h>
#include <math.h>

typedef __attribute__((ext_vector_type(2))) float v2f;
typedef __attribute__((ext_vector_type(8))) float v8f;

#define NPTS   4096
#define NBATCH 16

// Each wave owns a 16-row strip of the "row" point set and sweeps all 4096
// "column" points in 16-wide tiles. One V_WMMA_F32_16X16X4_F32 per tile
// computes e[m][n] = t2[n] - 2*<p_m, t_n> directly by packing:
//   A[m][0..2] = -2 * p_m,  A[m][3] = 1
//   B[0..2][n] =  t_n,      B[3][n] = |t_n|^2
//   C = 0
// Row-min of e is tracked per lane slot; sqrt(max(0, p2 + min)) applied once
// at the end (monotone, so it commutes with the min).
__global__ __launch_bounds__(256) void chamfer_rowmin_kernel(
    const float* __restrict__ rows,   // (NBATCH, NPTS, 3) — points owning the min
    const float* __restrict__ cols,   // (NBATCH, NPTS, 3) — points minimized over
    float* __restrict__ rowMinOut)    // (NBATCH, NPTS) nearest-neighbor distance
{
    const int lane = threadIdx.x & 31;
    const int wave = threadIdx.x >> 5;           // 0..7
    const int b    = blockIdx.y;
    const int m0   = blockIdx.x * 128 + wave * 16;
    const bool hi  = lane >= 16;
    const int  l16 = lane & 15;

    const float* __restrict__ Pb = rows + (size_t)b * NPTS * 3;
    const float* __restrict__ Tb = cols + (size_t)b * NPTS * 3;

    // ---- A fragment (16x4), rows scaled by -2, K=3 slot = 1.0 ----
    // VGPR0: K=0 (lanes 0-15) / K=2 (lanes 16-31); VGPR1: K=1 / K=3.
    const int mA = m0 + l16;
    const float ax = Pb[mA * 3 + 0];
    const float ay = Pb[mA * 3 + 1];
    const float az = Pb[mA * 3 + 2];
    v2f afrag;
    afrag.x = -2.0f * (hi ? az : ax);
    afrag.y = hi ? 1.0f : (-2.0f * ay);

    // ---- |p|^2 for the 8 rows this lane's D slots map to: m = m0 + j + 8*hi ----
    float p2r[8];
#pragma unroll
    for (int j = 0; j < 8; ++j) {
        const int m = m0 + j + (hi ? 8 : 0);
        const float x = Pb[m * 3 + 0];
        const float y = Pb[m * 3 + 1];
        const float z = Pb[m * 3 + 2];
        p2r[j] = x * x + y * y + z * z;
    }

    float rmin[8];
#pragma unroll
    for (int j = 0; j < 8; ++j) rmin[j] = 3.0e38f;

    const v8f czero = {};   // loop-invariant zero accumulator input

    // ---- Sweep all columns in 16-wide tiles ----
    for (int n0 = 0; n0 < NPTS; n0 += 16) {
        const int n = n0 + l16;
        const float tx = Tb[n * 3 + 0];
        const float ty = Tb[n * 3 + 1];
        const float tz = Tb[n * 3 + 2];
        const float t2 = tx * tx + ty * ty + tz * tz;

        // B fragment (4x16): VGPR0: K=0 / K=2; VGPR1: K=1 / K=3(=t2).
        v2f bfrag;
        bfrag.x = hi ? tz : tx;
        bfrag.y = hi ? t2 : ty;

        // D = (-2P | 1) x (T | t2) + 0  ==  t2[n] - 2*<p,t>
        v8f d = __builtin_amdgcn_wmma_f32_16x16x4_f32(
            /*neg_a=*/false, afrag, /*neg_b=*/false, bfrag,
            /*c_mod=*/(short)0, czero, /*reuse_a=*/false, /*reuse_b=*/false);

#pragma unroll
        for (int j = 0; j < 8; ++j) rmin[j] = fminf(rmin[j], d[j]);
    }

    // ---- Reduce mins across the 16 lanes of each half-wave ----
#pragma unroll
    for (int j = 0; j < 8; ++j) {
        float v = rmin[j];
        v = fminf(v, __shfl_xor(v, 1, 16));
        v = fminf(v, __shfl_xor(v, 2, 16));
        v = fminf(v, __shfl_xor(v, 4, 16));
        v = fminf(v, __shfl_xor(v, 8, 16));
        rmin[j] = v;
    }

    // Lane 0 writes rows m0+0..7, lane 16 writes rows m0+8..15.
    if (l16 == 0) {
#pragma unroll
        for (int j = 0; j < 8; ++j) {
            const float d2   = p2r[j] + rmin[j];
            const float dist = sqrtf(fmaxf(d2, 0.0f));
            rowMinOut[(size_t)b * NPTS + m0 + j + (hi ? 8 : 0)] = dist;
        }
    }
}

// Sum both min arrays (2 * NBATCH * NPTS floats), apply mean over batch and weight.
__global__ __launch_bounds__(256) void chamfer_reduce_kernel(
    const float* __restrict__ mins, float* __restrict__ out, int n)
{
    __shared__ float sdata[256];
    float s = 0.0f;
    for (int i = threadIdx.x; i < n; i += 256) s += mins[i];
    sdata[threadIdx.x] = s;
    __syncthreads();
    for (int off = 128; off > 0; off >>= 1) {
        if ((int)threadIdx.x < off) sdata[threadIdx.x] += sdata[threadIdx.x + off];
        __syncthreads();
    }
    if (threadIdx.x == 0) out[0] = 0.01f * sdata[0] / (float)NBATCH;
}

extern "C" void kernel_launch(void* const* d_in, const int* in_sizes, int n_in,
                              void* d_out, int out_size, void* d_ws, size_t ws_size,
                              hipStream_t stream) {
    (void)in_sizes; (void)n_in; (void)out_size; (void)ws_size;

    const float* shape = (const float*)d_in[0];   // (16, 4096, 3) f32
    const float* tmpl  = (const float*)d_in[1];   // (16, 4096, 3) f32

    float* minP = (float*)d_ws;                   // (16, 4096): min over template per shape pt
    float* minT = minP + (size_t)NBATCH * NPTS;   // (16, 4096): min over shape per template pt

    dim3 grid(NPTS / 128, NBATCH);  // 8 waves/block, 16 rows/wave -> 128 rows/block
    dim3 block(256);

    // Pass 1: rows = shape, cols = template  -> min_p2t
    chamfer_rowmin_kernel<<<grid, block, 0, stream>>>(shape, tmpl, minP);
    // Pass 2: rows = template, cols = shape  -> min_t2p (transposed roles)
    chamfer_rowmin_kernel<<<grid, block, 0, stream>>>(tmpl, shape, minT);

    chamfer_reduce_kernel<<<1, 256, 0, stream>>>((const float*)d_ws, (float*)d_out,
                                                 2 * NBATCH * NPTS);
}